// GConv_87780541595781
// MI455X (gfx1250) — compile-verified
//
#include <hip/hip_runtime.h>
#include <hip/hip_bf16.h>

typedef __attribute__((ext_vector_type(2))) float v2f;
typedef __attribute__((ext_vector_type(8))) float v8f;

#define D 128
// float2 units per K-pair row: 128 cols + 16 pad -> 288 words, 288 mod 64 = 32,
// so kpair-0 lanes (0-15) and kpair-1 lanes (16-31) hit disjoint bank halves.
#define PAIR_STRIDE 144

// ---------------------------------------------------------------------------
// deg[i] = 1.0 (self-loop weight)
__global__ __launch_bounds__(256) void fill_one_kernel(float* __restrict__ p, int n) {
    int i = blockIdx.x * blockDim.x + threadIdx.x;
    if (i < n) p[i] = 1.0f;
}

// deg[dst[e]] += w[e]
__global__ __launch_bounds__(256) void deg_kernel(const int* __restrict__ dst,
                                                  const float* __restrict__ w,
                                                  float* __restrict__ deg, int ne) {
    int e = blockIdx.x * blockDim.x + threadIdx.x;
    if (e < ne) atomicAdd(&deg[dst[e]], w[e]);
}

// dis = deg > 0 ? rsqrt(max(deg,1e-12)) : 0   (in place)
__global__ __launch_bounds__(256) void rsqrt_kernel(float* __restrict__ p, int n) {
    int i = blockIdx.x * blockDim.x + threadIdx.x;
    if (i < n) {
        float d = p[i];
        p[i] = (d > 0.0f) ? __frsqrt_rn(fmaxf(d, 1e-12f)) : 0.0f;
    }
}

// zero n float4's
__global__ __launch_bounds__(256) void zero_kernel(float4* __restrict__ p, int n4) {
    int i = blockIdx.x * blockDim.x + threadIdx.x;
    if (i < n4) p[i] = make_float4(0.f, 0.f, 0.f, 0.f);
}

// ---------------------------------------------------------------------------
// C[N,128] = A[N,128] @ W[128,128] via V_WMMA_F32_16X16X4_F32.
// One 256-thread block (8 waves) -> 128 rows; each wave owns a 16x128 strip.
// W is staged in LDS pre-swizzled as (K-pair, col) float2 records so every
// B fragment is a single aligned ds_load_b64.
__global__ __launch_bounds__(256) void gemm128_kernel(const float* __restrict__ A,
                                                      const float* __restrict__ W,
                                                      float* __restrict__ C, int nrows) {
    __shared__ v2f sB[64 * PAIR_STRIDE];   // 64 K-pairs x (128 cols + pad), 73.7 KB

    int tid = threadIdx.x;
    // Stage: sB[p*PAIR_STRIDE + col] = (W[2p, col], W[2p+1, col]).
    // 64 pairs x 32 col-quads = 2048 items, 8 iterations of 256 threads.
    for (int it = tid; it < 64 * 32; it += 256) {
        int p = it >> 5;
        int c = (it & 31) * 4;
        float4 va = *(const float4*)(W + (size_t)(2 * p) * D + c);
        float4 vb = *(const float4*)(W + (size_t)(2 * p + 1) * D + c);
        v2f* dst = &sB[p * PAIR_STRIDE + c];
        v2f t0; t0.x = va.x; t0.y = vb.x;
        v2f t1; t1.x = va.y; t1.y = vb.y;
        v2f t2; t2.x = va.z; t2.y = vb.z;
        v2f t3; t3.x = va.w; t3.y = vb.w;
        dst[0] = t0; dst[1] = t1; dst[2] = t2; dst[3] = t3;
    }
    __syncthreads();

    int wave = tid >> 5;
    int lane = tid & 31;
    int rowTile = blockIdx.x * 8 + wave;        // 16 rows per tile
    if (rowTile * 16 >= nrows) return;          // wave-uniform (nrows % 16 == 0)

    int lane15 = lane & 15;
    int kpair  = lane >> 4;                     // 0 or 1
    int kLane  = kpair * 2;                     // K sub-offset: 0 or 2
    const float* Arow = A + (size_t)(rowTile * 16 + lane15) * D;

    v8f acc[8] = {};                            // 8 column tiles of 16

    for (int kb = 0; kb < D; kb += 4) {
        // A fragment 16x4: lane holds (row = lane&15, k = kb+kLane, kb+kLane+1)
        v2f a = *(const v2f*)(Arow + kb + kLane);
        // B fragments: one aligned float2 LDS load each, adjacent j 128B apart.
        const v2f* brow = &sB[((kb >> 1) + kpair) * PAIR_STRIDE + lane15];
#pragma unroll
        for (int j = 0; j < 8; ++j) {
            v2f b = brow[j * 16];
            acc[j] = __builtin_amdgcn_wmma_f32_16x16x4_f32(
                false, a, false, b, (short)0, acc[j], false, false);
        }
    }

    // C tile layout: VGPR r -> lanes 0-15 row r, lanes 16-31 row r+8
    int rowHalf = kpair * 8;
#pragma unroll
    for (int j = 0; j < 8; ++j) {
#pragma unroll
        for (int r = 0; r < 8; ++r) {
            int row = rowTile * 16 + rowHalf + r;
            if (row < nrows)
                C[(size_t)row * D + j * 16 + lane15] = acc[j][r];
        }
    }
}

// ---------------------------------------------------------------------------
// One wave per edge: agg[dst] += h[src] * (dis[src]*w*dis[dst]); lane handles 4 feats.
__global__ __launch_bounds__(256) void agg_kernel(const float* __restrict__ h,
                                                  const int* __restrict__ src,
                                                  const int* __restrict__ dst,
                                                  const float* __restrict__ w,
                                                  const float* __restrict__ dis,
                                                  float* __restrict__ agg, int ne) {
    int gid  = blockIdx.x * blockDim.x + threadIdx.x;
    int e    = gid >> 5;
    int lane = gid & 31;
    if (e >= ne) return;
    int s = src[e];
    int t = dst[e];
    float norm = dis[s] * w[e] * dis[t];
    float4 v = ((const float4*)(h + (size_t)s * D))[lane];
    float* ap = agg + (size_t)t * D + lane * 4;
    atomicAdd(ap + 0, v.x * norm);
    atomicAdd(ap + 1, v.y * norm);
    atomicAdd(ap + 2, v.z * norm);
    atomicAdd(ap + 3, v.w * norm);
}

// ---------------------------------------------------------------------------
// out = prelu(agg + h*dis^2 + b, a)   — fuses self-loop message, bias, PReLU.
__global__ __launch_bounds__(256) void epi_kernel(const float* __restrict__ agg,
                                                  const float* __restrict__ h,
                                                  const float* __restrict__ dis,
                                                  const float* __restrict__ bias,
                                                  const float* __restrict__ aprelu,
                                                  float* __restrict__ out, int n) {
    int idx = blockIdx.x * blockDim.x + threadIdx.x;   // one float4 each
    if (idx >= n * (D / 4)) return;
    int node = idx >> 5;
    int c    = (idx & 31) * 4;
    float ds = dis[node];
    float sl = ds * ds;
    float4 hv = *(const float4*)(h   + (size_t)node * D + c);
    float4 av = *(const float4*)(agg + (size_t)node * D + c);
    float4 bv = *(const float4*)(bias + c);
    float4 pv = *(const float4*)(aprelu + c);
    float4 z;
    z.x = fmaf(hv.x, sl, av.x) + bv.x;
    z.y = fmaf(hv.y, sl, av.y) + bv.y;
    z.z = fmaf(hv.z, sl, av.z) + bv.z;
    z.w = fmaf(hv.w, sl, av.w) + bv.w;
    z.x = (z.x >= 0.f) ? z.x : pv.x * z.x;
    z.y = (z.y >= 0.f) ? z.y : pv.y * z.y;
    z.z = (z.z >= 0.f) ? z.z : pv.z * z.z;
    z.w = (z.w >= 0.f) ? z.w : pv.w * z.w;
    *(float4*)(out + (size_t)node * D + c) = z;
}

// ---------------------------------------------------------------------------
extern "C" void kernel_launch(void* const* d_in, const int* in_sizes, int n_in,
                              void* d_out, int out_size, void* d_ws, size_t ws_size,
                              hipStream_t stream) {
    const float* x   = (const float*)d_in[0];
    const int*   ei  = (const int*)  d_in[1];   // [2,E]: src row then dst row
    const float* ew  = (const float*)d_in[2];
    const float* W1  = (const float*)d_in[3];
    const float* b1  = (const float*)d_in[4];
    const float* W2  = (const float*)d_in[5];
    const float* b2  = (const float*)d_in[6];
    const float* ap  = (const float*)d_in[7];

    const int N = in_sizes[0] / D;              // 100000
    const int E = in_sizes[2];                  // 1600000
    const int* src = ei;
    const int* dst = ei + E;

    float* out  = (float*)d_out;                // agg / z buffer (51.2 MB)
    float* dis  = (float*)d_ws;                 // N floats (deg -> dis in place)
    float* hbuf = dis + N;                      // N*128 floats (h1 then h2)

    const int nf4     = N * (D / 4);            // float4 count of a feature matrix
    const int gemmBlk = (N + 127) / 128;
    const int aggBlk  = (int)(((long long)E * 32 + 255) / 256);

    // --- normalization ---
    hipLaunchKernelGGL(fill_one_kernel, dim3((N + 255) / 256), dim3(256), 0, stream, dis, N);
    hipLaunchKernelGGL(deg_kernel,      dim3((E + 255) / 256), dim3(256), 0, stream, dst, ew, dis, E);
    hipLaunchKernelGGL(rsqrt_kernel,    dim3((N + 255) / 256), dim3(256), 0, stream, dis, N);

    // --- layer 1 ---
    hipLaunchKernelGGL(gemm128_kernel, dim3(gemmBlk), dim3(256), 0, stream, x, W1, hbuf, N);
    hipLaunchKernelGGL(zero_kernel,    dim3((nf4 + 255) / 256), dim3(256), 0, stream, (float4*)out, nf4);
    hipLaunchKernelGGL(agg_kernel,     dim3(aggBlk), dim3(256), 0, stream, hbuf, src, dst, ew, dis, out, E);
    hipLaunchKernelGGL(epi_kernel,     dim3((nf4 + 255) / 256), dim3(256), 0, stream, out, hbuf, dis, b1, ap, out, N);

    // --- layer 2 ---
    hipLaunchKernelGGL(gemm128_kernel, dim3(gemmBlk), dim3(256), 0, stream, out, W2, hbuf, N);
    hipLaunchKernelGGL(zero_kernel,    dim3((nf4 + 255) / 256), dim3(256), 0, stream, (float4*)out, nf4);
    hipLaunchKernelGGL(agg_kernel,     dim3(aggBlk), dim3(256), 0, stream, hbuf, src, dst, ew, dis, out, E);
    hipLaunchKernelGGL(epi_kernel,     dim3((nf4 + 255) / 256), dim3(256), 0, stream, out, hbuf, dis, b2, ap, out, N);
}